// FeatureNet_58231166599608
// MI455X (gfx1250) — compile-verified
//
#include <hip/hip_runtime.h>
#include <math.h>

typedef __attribute__((ext_vector_type(2))) float v2f;
typedef __attribute__((ext_vector_type(8))) float v8f;

#define K_VOX   20000
#define T_PTS   35
#define C_IN    7
#define NROW    700000      /* K_VOX * T_PTS  (== 16 * 43750 exactly) */
#define C1      16
#define C2IN    32
#define C2      64
#define CF      138
#define NTILE_M 43750       /* NROW / 16 */
#define EPSV    1e-5f

// ---------------------------------------------------------------------------
// Zero helpers
// ---------------------------------------------------------------------------
__global__ void zero_stats_kernel(float* __restrict__ stats) {
    stats[threadIdx.x] = 0.f;
}

__global__ __launch_bounds__(256)
void zero4_kernel(float4* __restrict__ p, long long n4) {
    long long i = (long long)blockIdx.x * blockDim.x + threadIdx.x;
    long long stride = (long long)gridDim.x * blockDim.x;
    float4 z = make_float4(0.f, 0.f, 0.f, 0.f);
    for (; i < n4; i += stride) p[i] = z;
}

// ---------------------------------------------------------------------------
// Layer 1: h1 = relu(feature @ w1 + b1), mask = (max_c feature != 0)
// Accumulates global sum / sumsq per channel for batch-norm stats.
// K=7 is far too small for WMMA; plain FMA, memory-bound anyway.
// ---------------------------------------------------------------------------
__global__ __launch_bounds__(256)
void l1_kernel(const float* __restrict__ feat, const float* __restrict__ w1,
               const float* __restrict__ b1, float* __restrict__ h1,
               float* __restrict__ maskv, float* __restrict__ sum1,
               float* __restrict__ sq1)
{
    __shared__ float s_sum[C1], s_sq[C1];
    int tid = threadIdx.x;
    if (tid < C1) { s_sum[tid] = 0.f; s_sq[tid] = 0.f; }
    __syncthreads();

    int row = blockIdx.x * 256 + tid;
    float h[C1];
#pragma unroll
    for (int u = 0; u < C1; ++u) h[u] = 0.f;

    if (row < NROW) {
        const float* f = feat + (size_t)row * C_IN;
        float x[C_IN];
#pragma unroll
        for (int c = 0; c < C_IN; ++c) x[c] = f[c];
        float vmax = x[0];
#pragma unroll
        for (int c = 1; c < C_IN; ++c) vmax = fmaxf(vmax, x[c]);
        maskv[row] = (vmax != 0.f) ? 1.f : 0.f;
#pragma unroll
        for (int u = 0; u < C1; ++u) {
            float acc = b1[u];
#pragma unroll
            for (int c = 0; c < C_IN; ++c) acc = fmaf(x[c], w1[c * C1 + u], acc);
            h[u] = fmaxf(acc, 0.f);
        }
        float4* dst = (float4*)(h1 + (size_t)row * C1);
#pragma unroll
        for (int q = 0; q < 4; ++q)
            dst[q] = make_float4(h[4*q+0], h[4*q+1], h[4*q+2], h[4*q+3]);
    }

    // wave32 butterfly reduction, then LDS, then 1 global atomic per channel/block
#pragma unroll
    for (int u = 0; u < C1; ++u) {
        float s = h[u];
        float q = h[u] * h[u];
#pragma unroll
        for (int off = 16; off; off >>= 1) {
            s += __shfl_xor(s, off, 32);
            q += __shfl_xor(q, off, 32);
        }
        if ((tid & 31) == 0) { atomicAdd(&s_sum[u], s); atomicAdd(&s_sq[u], q); }
    }
    __syncthreads();
    if (tid < C1) { atomicAdd(&sum1[tid], s_sum[tid]); atomicAdd(&sq1[tid], s_sq[tid]); }
}

// ---------------------------------------------------------------------------
// Finalize BN stats -> scale/shift (generic for layer1/layer2)
// ---------------------------------------------------------------------------
__global__ void finalize_kernel(float* __restrict__ stats,
                                const float* __restrict__ g,
                                const float* __restrict__ be,
                                int sumOff, int sqOff, int scOff, int shOff,
                                float invN)
{
    int c = threadIdx.x;
    float mean = stats[sumOff + c] * invN;
    float var  = stats[sqOff + c] * invN - mean * mean;
    float sc   = g[c] * rsqrtf(var + EPSV);
    stats[scOff + c] = sc;
    stats[shOff + c] = be[c] - mean * sc;
}

// ---------------------------------------------------------------------------
// BN1 + per-voxel max aggregate + concat + mask -> cat1 (K*T, 32)
// ---------------------------------------------------------------------------
__global__ __launch_bounds__(64)
void bncat1_kernel(const float* __restrict__ h1, const float* __restrict__ maskv,
                   const float* __restrict__ stats, float* __restrict__ cat1)
{
    __shared__ float pw[T_PTS][C1];
    __shared__ float agg[C1];
    int k = blockIdx.x;
    int tid = threadIdx.x;
    const float* sc = stats + 160;
    const float* sh = stats + 176;

    for (int idx = tid; idx < T_PTS * C1; idx += 64) {
        int t = idx >> 4, c = idx & 15;
        pw[t][c] = h1[((size_t)(k * T_PTS + t)) * C1 + c] * sc[c] + sh[c];
    }
    __syncthreads();
    if (tid < C1) {
        float m = pw[0][tid];
        for (int t = 1; t < T_PTS; ++t) m = fmaxf(m, pw[t][tid]);
        agg[tid] = m;
    }
    __syncthreads();
    for (int idx = tid; idx < T_PTS * C1; idx += 64) {
        int t = idx >> 4, c = idx & 15;
        float mm = maskv[k * T_PTS + t];
        size_t ro = ((size_t)(k * T_PTS + t)) * C2IN;
        cat1[ro + c]      = pw[t][c] * mm;
        cat1[ro + C1 + c] = agg[c]   * mm;
    }
}

// ---------------------------------------------------------------------------
// Layer 2 GEMM via f32 WMMA: h2 = relu(cat1 @ w2 + b2), + BN2 stats accumulate
// One wave computes a 16x64 output tile: 4 N-tiles x 8 chained 16x16x4 WMMAs.
// A fragment (16x4 f32): lanes 0-15 hold K={4s,4s+1}, lanes 16-31 K={4s+2,4s+3}.
// B fragment (4x16 f32): mirrored with N across lanes.
// ---------------------------------------------------------------------------
__global__ __launch_bounds__(256)
void l2_wmma_kernel(const float* __restrict__ cat1, const float* __restrict__ w2,
                    const float* __restrict__ b2, float* __restrict__ h2,
                    float* __restrict__ sum2, float* __restrict__ sq2)
{
    __shared__ float s_w2[C2IN * C2];   // 8 KB weight stage
    __shared__ float s_sum[C2], s_sq[C2];
    int tid = threadIdx.x;
    if (tid < C2) { s_sum[tid] = 0.f; s_sq[tid] = 0.f; }
    for (int i = tid; i < C2IN * C2; i += 256) s_w2[i] = w2[i];
    __syncthreads();

    int wave  = blockIdx.x * 8 + (tid >> 5);
    int lane  = tid & 31;
    int hsel  = lane >> 4;     // 0 -> K pair {0,1}, 1 -> K pair {2,3} within step
    int lan16 = lane & 15;

    if (wave < NTILE_M) {                       // wave-uniform: EXEC stays full
        size_t rowBase = (size_t)wave * 16;
        const float* arow = cat1 + (rowBase + lan16) * C2IN + 2 * hsel;
        v2f a[8];
#pragma unroll
        for (int s = 0; s < 8; ++s) a[s] = *(const v2f*)(arow + 4 * s);

#pragma unroll
        for (int n = 0; n < 4; ++n) {
            int ncol = n * 16 + lan16;
            float bias = b2[ncol];
            v8f acc;
#pragma unroll
            for (int v = 0; v < 8; ++v) acc[v] = bias;

#pragma unroll
            for (int s = 0; s < 8; ++s) {
                v2f bb;
                bb.x = s_w2[(4 * s + 2 * hsel)     * C2 + ncol];
                bb.y = s_w2[(4 * s + 2 * hsel + 1) * C2 + ncol];
                acc = __builtin_amdgcn_wmma_f32_16x16x4_f32(
                          false, a[s], false, bb, (short)0, acc, false, false);
            }

            float ls = 0.f, lq = 0.f;
#pragma unroll
            for (int v = 0; v < 8; ++v) {
                float hv = fmaxf(acc[v], 0.f);
                int m = v + 8 * hsel;
                h2[(rowBase + m) * C2 + ncol] = hv;
                ls += hv; lq += hv * hv;
            }
            // lanes L and L+16 share the same output channel
            ls += __shfl_xor(ls, 16, 32);
            lq += __shfl_xor(lq, 16, 32);
            if (hsel == 0) { atomicAdd(&s_sum[ncol], ls); atomicAdd(&s_sq[ncol], lq); }
        }
    }
    __syncthreads();
    if (tid < C2) { atomicAdd(&sum2[tid], s_sum[tid]); atomicAdd(&sq2[tid], s_sq[tid]); }
}

// ---------------------------------------------------------------------------
// BN2 + aggregates + voxel max + histogram + scatter-add into grid
// 64 threads per voxel: thread c owns output channels c and 64+c.
// ---------------------------------------------------------------------------
__global__ __launch_bounds__(64)
void voxel_final_kernel(const float* __restrict__ feat, const int* __restrict__ coord,
                        const float* __restrict__ h2, const float* __restrict__ maskv,
                        const float* __restrict__ stats, float* __restrict__ grid)
{
    __shared__ float fin[CF + 6];
    __shared__ float s_hist[10];
    int k = blockIdx.x;
    int tid = threadIdx.x;
    if (tid < 10) s_hist[tid] = 0.f;
    __syncthreads();

    const float* sc2 = stats + 192;
    const float* sh2 = stats + 256;
    float scc = sc2[tid];
    float shh = sh2[tid];

    float mx1 = -INFINITY;   // max_t( pw2 * mask )
    float agg = -INFINITY;   // max_t( pw2 )         (pre-mask, per reference)
    bool anyOne = false, anyZero = false;
    for (int t = 0; t < T_PTS; ++t) {
        float m  = maskv[k * T_PTS + t];
        float pw = h2[((size_t)(k * T_PTS + t)) * C2 + tid] * scc + shh;
        agg = fmaxf(agg, pw);
        mx1 = fmaxf(mx1, pw * m);
        anyOne  |= (m != 0.f);
        anyZero |= (m == 0.f);
    }
    // max_t( agg * mask[t] ): values are agg (where mask==1) or 0 (where mask==0)
    float mx2 = anyOne ? (anyZero ? fmaxf(agg, 0.f) : agg) : 0.f;
    fin[tid]      = mx1;
    fin[C2 + tid] = mx2;

    if (tid < T_PTS) {
        float v = feat[((size_t)(k * T_PTS + tid)) * C_IN + 3];
        int idx = (int)floorf(v * 10.f);
        idx = min(max(idx, 0), 9);
        float valid = (v >= 0.f && v <= 1.f) ? 1.f : 0.f;
        atomicAdd(&s_hist[idx], valid);
    }
    __syncthreads();
    if (tid < 10) fin[128 + tid] = s_hist[tid];
    __syncthreads();

    const int* c = coord + k * 4;
    long long flat = ((((long long)c[0] * 10 + c[1]) * 400 + c[2]) * 352 + c[3]);
    float* dst = grid + flat * CF;
    for (int j = tid; j < CF; j += 64) atomicAdd(&dst[j], fin[j]);
}

// ---------------------------------------------------------------------------
// Host launcher
// ---------------------------------------------------------------------------
extern "C" void kernel_launch(void* const* d_in, const int* in_sizes, int n_in,
                              void* d_out, int out_size, void* d_ws, size_t ws_size,
                              hipStream_t stream)
{
    (void)in_sizes; (void)n_in; (void)ws_size;
    const float* feat  = (const float*)d_in[0];
    const int*   coord = (const int*)  d_in[1];
    const float* w1    = (const float*)d_in[2];
    const float* b1    = (const float*)d_in[3];
    const float* g1    = (const float*)d_in[4];
    const float* be1   = (const float*)d_in[5];
    const float* w2    = (const float*)d_in[6];
    const float* b2    = (const float*)d_in[7];
    const float* g2    = (const float*)d_in[8];
    const float* be2   = (const float*)d_in[9];
    float* out = (float*)d_out;

    // Workspace layout (floats):
    // [0,512)   stats: sum1(16) sq1(16) sum2(64) sq2(64) sc1@160 sh1@176 sc2@192 sh2@256
    // mask:  NROW            (2.8 MB)
    // h1:    NROW*16         (44.8 MB)
    // cat1:  NROW*32         (89.6 MB)
    // h2:    NROW*64         (179.2 MB)
    float* ws    = (float*)d_ws;
    float* stats = ws;
    float* maskv = ws + 512;
    float* h1    = maskv + (size_t)NROW;
    float* cat1  = h1    + (size_t)NROW * C1;
    float* h2    = cat1  + (size_t)NROW * C2IN;

    float* sum1 = stats + 0;
    float* sq1  = stats + 16;
    float* sum2 = stats + 32;
    float* sq2  = stats + 96;

    hipLaunchKernelGGL(zero_stats_kernel, dim3(1), dim3(512), 0, stream, stats);

    long long n4 = (long long)out_size / 4;   // 194,304,000 / 4
    hipLaunchKernelGGL(zero4_kernel, dim3(4096), dim3(256), 0, stream,
                       (float4*)out, n4);

    hipLaunchKernelGGL(l1_kernel, dim3((NROW + 255) / 256), dim3(256), 0, stream,
                       feat, w1, b1, h1, maskv, sum1, sq1);

    hipLaunchKernelGGL(finalize_kernel, dim3(1), dim3(C1), 0, stream,
                       stats, g1, be1, 0, 16, 160, 176, 1.0f / (float)NROW);

    hipLaunchKernelGGL(bncat1_kernel, dim3(K_VOX), dim3(64), 0, stream,
                       h1, maskv, stats, cat1);

    hipLaunchKernelGGL(l2_wmma_kernel, dim3((NTILE_M + 7) / 8), dim3(256), 0, stream,
                       cat1, w2, b2, h2, sum2, sq2);

    hipLaunchKernelGGL(finalize_kernel, dim3(1), dim3(C2), 0, stream,
                       stats, g2, be2, 32, 96, 192, 256, 1.0f / (float)NROW);

    hipLaunchKernelGGL(voxel_final_kernel, dim3(K_VOX), dim3(64), 0, stream,
                       feat, coord, h2, maskv, stats, out);
}